// ResBlock_47064251630157
// MI455X (gfx1250) — compile-verified
//
#include <hip/hip_runtime.h>

// MI455X / gfx1250 implementation of the GraphConv ResBlock.
//
// Roofline: GEMMs total ~11.5 GFLOP fp32 (trivial for WMMA); cost is the two
// segment_sums (~1.2 GB of edge-feature gather + ~300M f32 atomics), which are
// L2-resident (x=25.6MB, y=51.2MB << 192MB L2). So: edge-parallel float4
// gathers + global_atomic_add_f32 scatters, and precision-preserving fp32
// WMMA (v_wmma_f32_16x16x4_f32) for all five GEMMs, fused with bias/LN/ReLU.

typedef __attribute__((ext_vector_type(2))) float v2f;
typedef __attribute__((ext_vector_type(8))) float v8f;

#define LN_EPS 1e-5f

// ---------------- zero fill (agg buffers must be zeroed every call) --------
__global__ void zero_f4(float4* __restrict__ p, long n4) {
  long i = (long)blockIdx.x * blockDim.x + threadIdx.x;
  if (i < n4) p[i] = make_float4(0.f, 0.f, 0.f, 0.f);
}

// ---------------- edge scatter-add: agg[dst] += feat[src] ------------------
// C = feature width (64 or 128). C/4 threads per edge, float4 gather,
// 4 x global_atomic_add_f32 scatter (L2-resident working set).
template <int C>
__global__ void scatter_add(const float* __restrict__ feat,
                            const int* __restrict__ srcv,
                            const int* __restrict__ dstv,
                            float* __restrict__ agg, int n_edges) {
  constexpr int PER = C / 4;
  long tid = (long)blockIdx.x * blockDim.x + threadIdx.x;
  int e = (int)(tid / PER);
  if (e >= n_edges) return;
  int c = (int)(tid % PER) * 4;
  int s = srcv[e];
  int d = dstv[e];
  const float4 v = *(const float4*)(feat + (long)s * C + c);
  float* o = agg + (long)d * C + c;
  atomicAdd(o + 0, v.x);
  atomicAdd(o + 1, v.y);
  atomicAdd(o + 2, v.z);
  atomicAdd(o + 3, v.w);
}

// ---------------- layer 0: y = relu(LN(x@Wr + agg@Wn + b)) -----------------
// Block = 256 threads = 8 waves; block computes 16 rows x 128 cols.
// Wave w owns the 16x16 tile at cols [16w,16w+16) using fp32 WMMA (K step 4).
// fp32 WMMA VGPR layout (ISA 7.12.2): lane = 16h + n,
//   A: a[i] = A[n][k0 + 2h + i];  B: b[i] = B[k0 + 2h + i][col];
//   D: acc[r] = D[r + 8h][n].
__global__ __launch_bounds__(256) void gemm_ln0(
    const float* __restrict__ x,       // [N,64]
    const float* __restrict__ agg,     // [N,64]
    const float* __restrict__ w_root,  // [64,128]
    const float* __restrict__ w_nbr,   // [64,128]
    const float* __restrict__ bias,    // [128]
    const float* __restrict__ gamma,   // [128]
    const float* __restrict__ beta,    // [128]
    float* __restrict__ y) {           // [N,128]
  __shared__ float Xs[16][68];   // stride 68 (=4 mod 64): conflict-free gather
  __shared__ float Gs[16][68];
  __shared__ float Zs[16][132];
  __shared__ float Ps[16][16];
  __shared__ float Qs[16][16];
  __shared__ float MuS[16], RsS[16];

  const int t = threadIdx.x;
  const long rowBase = (long)blockIdx.x * 16;

  // cooperative stage of the 16x64 X and AGG tiles (4 floats / thread each)
  {
    int idx = t * 4;
    int r = idx >> 6;
    int c = idx & 63;
    *(float4*)&Xs[r][c] = *(const float4*)(x + (rowBase + r) * 64 + c);
    *(float4*)&Gs[r][c] = *(const float4*)(agg + (rowBase + r) * 64 + c);
  }
  __syncthreads();

  const int lane = t & 31;
  const int h = lane >> 4;
  const int n = lane & 15;
  const int col = (t >> 5) * 16 + n;

  v8f acc = {0.f, 0.f, 0.f, 0.f, 0.f, 0.f, 0.f, 0.f};
#pragma unroll
  for (int k0 = 0; k0 < 64; k0 += 4) {
    const int ka = k0 + 2 * h;
    v2f a, b;
    a.x = Xs[n][ka];
    a.y = Xs[n][ka + 1];
    b.x = w_root[ka * 128 + col];
    b.y = w_root[(ka + 1) * 128 + col];
    acc = __builtin_amdgcn_wmma_f32_16x16x4_f32(false, a, false, b, (short)0,
                                                acc, false, false);
  }
#pragma unroll
  for (int k0 = 0; k0 < 64; k0 += 4) {
    const int ka = k0 + 2 * h;
    v2f a, b;
    a.x = Gs[n][ka];
    a.y = Gs[n][ka + 1];
    b.x = w_nbr[ka * 128 + col];
    b.y = w_nbr[(ka + 1) * 128 + col];
    acc = __builtin_amdgcn_wmma_f32_16x16x4_f32(false, a, false, b, (short)0,
                                                acc, false, false);
  }
  const float bv = bias[col];
#pragma unroll
  for (int r = 0; r < 8; ++r) Zs[r + 8 * h][col] = acc[r] + bv;
  __syncthreads();

  // LayerNorm over the 128 features of each of the 16 rows, then ReLU.
  {
    int r = t >> 4, seg = t & 15;
    float s = 0.f, q = 0.f;
#pragma unroll
    for (int j = 0; j < 8; ++j) {
      float v = Zs[r][seg * 8 + j];
      s += v;
      q += v * v;
    }
    Ps[r][seg] = s;
    Qs[r][seg] = q;
  }
  __syncthreads();
  if (t < 16) {
    float s = 0.f, q = 0.f;
#pragma unroll
    for (int j = 0; j < 16; ++j) {
      s += Ps[t][j];
      q += Qs[t][j];
    }
    float mu = s * (1.f / 128.f);
    float var = q * (1.f / 128.f) - mu * mu;
    MuS[t] = mu;
    RsS[t] = rsqrtf(var + LN_EPS);
  }
  __syncthreads();
  {
    int r = t >> 4, seg = t & 15;
    float mu = MuS[r], rs = RsS[r];
    float o[8];
#pragma unroll
    for (int j = 0; j < 8; ++j) {
      int c = seg * 8 + j;
      float v = (Zs[r][c] - mu) * rs * gamma[c] + beta[c];
      o[j] = v > 0.f ? v : 0.f;
    }
    float4* op = (float4*)(y + (rowBase + r) * 128 + seg * 8);
    op[0] = make_float4(o[0], o[1], o[2], o[3]);
    op[1] = make_float4(o[4], o[5], o[6], o[7]);
  }
}

// ---- layer 1: out = relu(LN(y@Wr1 + agg1@Wn1 + b1 + x@Wlin + blin)) -------
// `y` may alias `out` (same 16 rows only, staged to LDS before overwrite),
// so no __restrict__ on those two.
__global__ __launch_bounds__(256) void gemm_ln1(
    const float* y,                    // [N,128]
    const float* __restrict__ agg,     // [N,128]
    const float* __restrict__ x,       // [N,64]
    const float* __restrict__ w_root,  // [128,128]
    const float* __restrict__ w_nbr,   // [128,128]
    const float* __restrict__ w_lin,   // [64,128]
    const float* __restrict__ b1,      // [128]
    const float* __restrict__ b_lin,   // [128]
    const float* __restrict__ gamma,   // [128]
    const float* __restrict__ beta,    // [128]
    float* out) {                      // [N,128]
  __shared__ float Ys[16][132];
  __shared__ float Gs[16][132];
  __shared__ float Xs[16][68];
  __shared__ float Zs[16][132];
  __shared__ float Ps[16][16];
  __shared__ float Qs[16][16];
  __shared__ float MuS[16], RsS[16];

  const int t = threadIdx.x;
  const long rowBase = (long)blockIdx.x * 16;

  {  // stage Y and AGG1 tiles (16x128 each: 8 floats / thread)
    int idx = t * 8;
    int r = idx >> 7;
    int c = idx & 127;
    *(float4*)&Ys[r][c] = *(const float4*)(y + (rowBase + r) * 128 + c);
    *(float4*)&Ys[r][c + 4] = *(const float4*)(y + (rowBase + r) * 128 + c + 4);
    *(float4*)&Gs[r][c] = *(const float4*)(agg + (rowBase + r) * 128 + c);
    *(float4*)&Gs[r][c + 4] =
        *(const float4*)(agg + (rowBase + r) * 128 + c + 4);
  }
  {  // stage X tile (16x64: 4 floats / thread)
    int idx = t * 4;
    int r = idx >> 6;
    int c = idx & 63;
    *(float4*)&Xs[r][c] = *(const float4*)(x + (rowBase + r) * 64 + c);
  }
  __syncthreads();

  const int lane = t & 31;
  const int h = lane >> 4;
  const int n = lane & 15;
  const int col = (t >> 5) * 16 + n;

  v8f acc = {0.f, 0.f, 0.f, 0.f, 0.f, 0.f, 0.f, 0.f};
#pragma unroll
  for (int k0 = 0; k0 < 128; k0 += 4) {
    const int ka = k0 + 2 * h;
    v2f a, b;
    a.x = Ys[n][ka];
    a.y = Ys[n][ka + 1];
    b.x = w_root[ka * 128 + col];
    b.y = w_root[(ka + 1) * 128 + col];
    acc = __builtin_amdgcn_wmma_f32_16x16x4_f32(false, a, false, b, (short)0,
                                                acc, false, false);
  }
#pragma unroll
  for (int k0 = 0; k0 < 128; k0 += 4) {
    const int ka = k0 + 2 * h;
    v2f a, b;
    a.x = Gs[n][ka];
    a.y = Gs[n][ka + 1];
    b.x = w_nbr[ka * 128 + col];
    b.y = w_nbr[(ka + 1) * 128 + col];
    acc = __builtin_amdgcn_wmma_f32_16x16x4_f32(false, a, false, b, (short)0,
                                                acc, false, false);
  }
#pragma unroll
  for (int k0 = 0; k0 < 64; k0 += 4) {  // residual projection x @ w_lin
    const int ka = k0 + 2 * h;
    v2f a, b;
    a.x = Xs[n][ka];
    a.y = Xs[n][ka + 1];
    b.x = w_lin[ka * 128 + col];
    b.y = w_lin[(ka + 1) * 128 + col];
    acc = __builtin_amdgcn_wmma_f32_16x16x4_f32(false, a, false, b, (short)0,
                                                acc, false, false);
  }
  const float bv = b1[col] + b_lin[col];
#pragma unroll
  for (int r = 0; r < 8; ++r) Zs[r + 8 * h][col] = acc[r] + bv;
  __syncthreads();

  {
    int r = t >> 4, seg = t & 15;
    float s = 0.f, q = 0.f;
#pragma unroll
    for (int j = 0; j < 8; ++j) {
      float v = Zs[r][seg * 8 + j];
      s += v;
      q += v * v;
    }
    Ps[r][seg] = s;
    Qs[r][seg] = q;
  }
  __syncthreads();
  if (t < 16) {
    float s = 0.f, q = 0.f;
#pragma unroll
    for (int j = 0; j < 16; ++j) {
      s += Ps[t][j];
      q += Qs[t][j];
    }
    float mu = s * (1.f / 128.f);
    float var = q * (1.f / 128.f) - mu * mu;
    MuS[t] = mu;
    RsS[t] = rsqrtf(var + LN_EPS);
  }
  __syncthreads();
  {
    int r = t >> 4, seg = t & 15;
    float mu = MuS[r], rs = RsS[r];
    float o[8];
#pragma unroll
    for (int j = 0; j < 8; ++j) {
      int c = seg * 8 + j;
      float v = (Zs[r][c] - mu) * rs * gamma[c] + beta[c];
      o[j] = v > 0.f ? v : 0.f;
    }
    float4* op = (float4*)(out + (rowBase + r) * 128 + seg * 8);
    op[0] = make_float4(o[0], o[1], o[2], o[3]);
    op[1] = make_float4(o[4], o[5], o[6], o[7]);
  }
}

extern "C" void kernel_launch(void* const* d_in, const int* in_sizes, int n_in,
                              void* d_out, int out_size, void* d_ws,
                              size_t ws_size, hipStream_t stream) {
  const float* x = (const float*)d_in[0];
  const int* ei = (const int*)d_in[1];
  const float* w_root0 = (const float*)d_in[2];
  const float* w_nbr0 = (const float*)d_in[3];
  const float* b0 = (const float*)d_in[4];
  const float* w_root1 = (const float*)d_in[5];
  const float* w_nbr1 = (const float*)d_in[6];
  const float* b1 = (const float*)d_in[7];
  const float* gamma0 = (const float*)d_in[8];
  const float* beta0 = (const float*)d_in[9];
  const float* gamma1 = (const float*)d_in[10];
  const float* beta1 = (const float*)d_in[11];
  const float* w_lin = (const float*)d_in[12];
  const float* b_lin = (const float*)d_in[13];

  const int n_nodes = in_sizes[0] / 64;   // 100000
  const int n_edges = in_sizes[1] / 2;    // 1600000
  const int* src = ei;
  const int* dst = ei + n_edges;

  float* agg0 = (float*)d_ws;                       // [N,64]
  float* agg1 = agg0 + (size_t)n_nodes * 64;        // [N,128]
  const size_t needY = (size_t)n_nodes * (64 + 128 + 128) * sizeof(float);
  // y intermediate: in ws if it fits, else reuse d_out (safe: gemm_ln1 stages
  // its own 16 rows to LDS before overwriting them).
  float* ybuf =
      (ws_size >= needY) ? agg1 + (size_t)n_nodes * 128 : (float*)d_out;
  float* out = (float*)d_out;

  // 1) zero agg0+agg1 (contiguous N*192 floats)
  long n4 = (long)n_nodes * 192 / 4;
  zero_f4<<<(unsigned)((n4 + 255) / 256), 256, 0, stream>>>((float4*)agg0, n4);

  // 2) layer-0 aggregation: agg0[dst] += x[src]
  long t0 = (long)n_edges * 16;
  scatter_add<64><<<(unsigned)((t0 + 255) / 256), 256, 0, stream>>>(
      x, src, dst, agg0, n_edges);

  // 3) y = relu(LN(x@Wr0 + agg0@Wn0 + b0))
  const int nb = n_nodes / 16;  // N is a multiple of 16
  gemm_ln0<<<nb, 256, 0, stream>>>(x, agg0, w_root0, w_nbr0, b0, gamma0, beta0,
                                   ybuf);

  // 4) layer-1 aggregation: agg1[dst] += y[src]
  long t1 = (long)n_edges * 32;
  scatter_add<128><<<(unsigned)((t1 + 255) / 256), 256, 0, stream>>>(
      ybuf, src, dst, agg1, n_edges);

  // 5) out = relu(LN(y@Wr1 + agg1@Wn1 + b1 + x@Wlin + blin))
  gemm_ln1<<<nb, 256, 0, stream>>>(ybuf, agg1, x, w_root1, w_nbr1, w_lin, b1,
                                   b_lin, gamma1, beta1, out);
}